// Freeness_59777354826389
// MI455X (gfx1250) — compile-verified
//
#include <hip/hip_runtime.h>

// DNC freeness / write-allocation kernel for gfx1250 (MI455X).
// One 1024-thread (32-wave) workgroup per batch row; 204KB dynamic LDS
// (CDNA5-only: 320KB per WGP). Hybrid register/LDS bitonic u64 sort
// reproduces jax.lax.top_k ordering (nonusage desc, index asc) exactly:
// strides >=16 run in LDS, strides <=8 run in VGPRs (wave32 v_cndmask
// networks). f32 hierarchical cumprod matches the reference's
// underflow-to-zero behavior. Output rows are streamed out with async
// LDS->global DMA (ASYNCcnt) overlapped with the next iteration's sort.

#ifndef USE_ASYNC_OUT
#define USE_ASYNC_OUT 1
#endif

typedef unsigned long long u64;

#define NM   16384            // memory slots per row
#define NT   1024             // threads per block
#define KPT  16               // NM / NT slots per thread
#define NW   4                // num_writes
#define NR   8                // num_reads
#define NB   256              // batch
#define EPSF 1e-6f
#define ARR_U64 (NM + NM / 16)  // padded sort array (1 u64 pad per 16 slots)

extern __shared__ unsigned char smem_raw[];

// padded physical index: kills bank conflicts on per-thread 16-slot blocks
__device__ __forceinline__ int ph(int pos) { return pos + (pos >> 4); }

__device__ __forceinline__ unsigned lds_off(const void* p) {
  // generic -> LDS(AS3) addrspacecast, then ptrtoint gives the LDS byte offset
  return (unsigned)(unsigned long long)
      (const __attribute__((address_space(3))) unsigned char*)p;
}

__device__ __forceinline__ void wait_async0() {
#if __has_builtin(__builtin_amdgcn_s_wait_asynccnt)
  __builtin_amdgcn_s_wait_asynccnt(0);
#else
  asm volatile("s_wait_asynccnt 0" ::: "memory");
#endif
}

#if USE_ASYNC_OUT
__device__ __forceinline__ void async_store_b128(void* gaddr, unsigned ldsoff) {
  // VGLOBAL: GLOBAL_STORE_ASYNC_FROM_LDS_B128 (per-lane 16B LDS -> memory)
  asm volatile("global_store_async_from_lds_b128 %0, %1, off"
               :: "v"((unsigned long long)(uintptr_t)gaddr), "v"(ldsoff)
               : "memory");
}
#endif

// total-order key: ascending u64 == (nonusage desc, index asc) == top_k order
__device__ __forceinline__ u64 make_key(float n, unsigned m) {
  unsigned bits = __float_as_uint(n);
  unsigned asc  = bits ^ (0x80000000u | (unsigned)((int)bits >> 31));
  return ((u64)(~asc) << 32) | (u64)m;
}
__device__ __forceinline__ float key_n(u64 v) {
  unsigned asc  = ~(unsigned)(v >> 32);
  unsigned bits = (asc & 0x80000000u) ? (asc ^ 0x80000000u) : ~asc;
  return __uint_as_float(bits);
}

__device__ __forceinline__ void ce_reg(u64& a, u64& b, bool up) {
  bool sw = up ? (a > b) : (a < b);
  u64 x = sw ? b : a;
  u64 y = sw ? a : b;
  a = x; b = y;
}

// bitonic merge of a 16-slot register block, uniform direction (strides 8..1)
__device__ __forceinline__ void merge16(u64 r[KPT], bool up) {
#pragma unroll
  for (int st = 8; st >= 1; st >>= 1) {
#pragma unroll
    for (int g = 0; g < 8; ++g) {
      int j = 2 * g - (g & (st - 1));
      ce_reg(r[j], r[j + st], up);
    }
  }
}

// full bitonic network stages size=2..16 on a 16-slot register block;
// directions for size<=8 are compile-time (local j), size=16 is uniform.
__device__ __forceinline__ void sort16(u64 r[KPT], bool up16) {
#pragma unroll
  for (int sz = 2; sz <= 8; sz <<= 1) {
#pragma unroll
    for (int st = sz >> 1; st >= 1; st >>= 1) {
#pragma unroll
      for (int g = 0; g < 8; ++g) {
        int j = 2 * g - (g & (st - 1));
        ce_reg(r[j], r[j + st], ((j & sz) == 0));
      }
    }
  }
  merge16(r, up16);
}

__global__ void __launch_bounds__(NT)
freeness_kernel(const float* __restrict__ pww,    // (B, W, M)
                const float* __restrict__ prw,    // (B, R, M)
                const float* __restrict__ pu,     // (B, M)
                const float* __restrict__ fg,     // (B, R)
                const float* __restrict__ wgates, // (B, W)
                float* __restrict__ out)          // (B, W, M)
{
  // LDS: [0,136K) padded u64 sort array; [136K,200K) f32 staging row;
  //      [200K,204K) f32 scan array.
  u64*   arr  = (u64*)smem_raw;
  float* fArr = (float*)(smem_raw + (size_t)ARR_U64 * 8);
  float* sp   = (float*)(smem_raw + (size_t)ARR_U64 * 8 + (size_t)NM * 4);

  const int b = blockIdx.x;
  const int t = threadIdx.x;
  const int base = t * KPT;           // this thread's sorted-array block

  float fgv[NR];
#pragma unroll
  for (int r = 0; r < NR; ++r) fgv[r] = fg[b * NR + r];
  float wgv[NW];
#pragma unroll
  for (int i = 0; i < NW; ++i) wgv[i] = wgates[b * NW + i];

  const float* pwwB = pww + (size_t)b * NW * NM;
  const float* prwB = prw + (size_t)b * NR * NM;
  const float* puB  = pu  + (size_t)b * NM;

  // ---- Phase 0: usage after write weights and read-free gating ----
  float us[KPT];
#pragma unroll
  for (int k = 0; k < KPT; ++k) {
    int m = k * NT + t;
    float wprod = 1.0f;
#pragma unroll
    for (int w = 0; w < NW; ++w) wprod *= (1.0f - pwwB[(size_t)w * NM + m]);
    float write_w = 1.0f - wprod;
    float u0 = puB[m];
    float usage = u0 + (1.0f - u0) * write_w;
    float phi = 1.0f;
#pragma unroll
    for (int r = 0; r < NR; ++r) phi *= (1.0f - fgv[r] * prwB[(size_t)r * NM + m]);
    us[k] = usage * phi;
  }

  float* outB = out + (size_t)b * NW * NM;

  for (int it = 0; it < NW; ++it) {
    // ---- (a) build keys into LDS (coalesced layout m = k*NT + t) ----
#pragma unroll
    for (int k = 0; k < KPT; ++k) {
      int m = k * NT + t;
      float u = EPSF + (1.0f - EPSF) * us[k];
      float n = 1.0f - u;  // nonusage, exactly as the reference computes it
      arr[ph(m)] = make_key(n, (unsigned)m);
    }
    __syncthreads();

    // ---- (b) hybrid bitonic sort ----
    u64 r[KPT];
#pragma unroll
    for (int k = 0; k < KPT; ++k) r[k] = arr[ph(base + k)];
    sort16(r, ((base & 16) == 0));          // stages size = 2..16, in VGPRs

    for (int size = 32; size <= NM; size <<= 1) {
      // publish block, run cross-block strides (>=16) in LDS
#pragma unroll
      for (int k = 0; k < KPT; ++k) arr[ph(base + k)] = r[k];
      for (int stride = size >> 1; stride >= 16; stride >>= 1) {
        __syncthreads();
#pragma unroll
        for (int p = 0; p < (NM / 2) / NT; ++p) {
          int gid = t + p * NT;
          int i = (gid << 1) - (gid & (stride - 1));
          int j = i + stride;
          bool up = ((i & size) == 0);
          u64 a = arr[ph(i)];
          u64 c = arr[ph(j)];
          bool sw = up ? (a > c) : (a < c);
          arr[ph(i)] = sw ? c : a;
          arr[ph(j)] = sw ? a : c;
        }
      }
      __syncthreads();
#pragma unroll
      for (int k = 0; k < KPT; ++k) r[k] = arr[ph(base + k)];
      merge16(r, ((base & size) == 0));     // strides 8..1, in VGPRs
    }
    // r[] now holds this thread's 16 consecutive globally-sorted slots.

    // ---- (c) exclusive cumprod of sorted usage (f32, matches reference
    // underflow): sequential over the register block, Hillis-Steele scan
    // over the 1024 segment products.
    float segp = 1.0f;
#pragma unroll
    for (int k = 0; k < KPT; ++k) segp *= (1.0f - key_n(r[k]));
    sp[t] = segp;
    __syncthreads();
    for (int d = 1; d < NT; d <<= 1) {
      float mine  = sp[t];
      float other = (t >= d) ? sp[t - d] : 1.0f;
      __syncthreads();
      sp[t] = mine * other;
      __syncthreads();
    }
    float running = (t == 0) ? 1.0f : sp[t - 1];

#if USE_ASYNC_OUT
    // previous iteration's output DMA must finish before fArr is reused;
    // each wave drains its own ASYNCcnt, the barrier makes it workgroup-wide.
    if (it > 0) wait_async0();
#endif
    __syncthreads();

    // ---- (d+e) alloc = nonusage * exclprod, scattered straight from regs ----
#pragma unroll
    for (int k = 0; k < KPT; ++k) {
      u64 v = r[k];
      unsigned idx = (unsigned)(v & 0xFFFFFFFFull);
      float n = key_n(v);
      fArr[idx] = n * running;
      running  *= (1.0f - n);
    }
    __syncthreads();

    // ---- (f) usage update from the scattered row ----
    float wgi = wgv[it];
    float* outIt = outB + (size_t)it * NM;
#pragma unroll
    for (int k = 0; k < KPT; ++k) {
      int m = k * NT + t;
      float a = fArr[m];
      us[k] = us[k] + (1.0f - us[k]) * wgi * a;
#if !USE_ASYNC_OUT
      outIt[m] = a;  // coalesced fallback store
#endif
    }
#if USE_ASYNC_OUT
    // DMA the finished 64KB row LDS -> global; overlaps the next sort.
#pragma unroll
    for (int q = 0; q < 4; ++q) {
      int e = t * 16 + q * 4;  // 16B-aligned, 4 floats per lane per op
      async_store_b128(outIt + e, lds_off(&fArr[e]));
    }
#endif
  }
#if USE_ASYNC_OUT
  wait_async0();
#endif
}

extern "C" void kernel_launch(void* const* d_in, const int* in_sizes, int n_in,
                              void* d_out, int out_size, void* d_ws, size_t ws_size,
                              hipStream_t stream) {
  (void)in_sizes; (void)n_in; (void)out_size; (void)d_ws; (void)ws_size;
  // setup_inputs order: inputs(unused), prev_write_weight, prev_read_weight,
  // prev_usage, free_gate, write_gates, num_writes(scalar, fixed at 4)
  const float* pww    = (const float*)d_in[1];
  const float* prw    = (const float*)d_in[2];
  const float* pu     = (const float*)d_in[3];
  const float* fg     = (const float*)d_in[4];
  const float* wgates = (const float*)d_in[5];
  float* out = (float*)d_out;

  const size_t shmem = (size_t)ARR_U64 * 8 + (size_t)NM * 4 + (size_t)NT * 4;
  // CDNA5 WGPs have 320KB LDS; opt in to >64KB dynamic LDS (deterministic,
  // non-stream call: safe under graph capture).
  hipFuncSetAttribute(reinterpret_cast<const void*>(freeness_kernel),
                      hipFuncAttributeMaxDynamicSharedMemorySize, (int)shmem);
  hipLaunchKernelGGL(freeness_kernel, dim3(NB), dim3(NT), shmem, stream,
                     pww, prw, pu, fg, wgates, out);
}